// RotationLayer_20538533609818
// MI455X (gfx1250) — compile-verified
//
#include <hip/hip_runtime.h>

// ---------------------------------------------------------------------------
// RotationLayer: out[b,n,:] = R @ x[b,n,:] - R @ x[b,n-1,:]  (out[b,0,:] = 0)
// R = [[c,-s],[s,c]], c/s from cos/sin(rand_value[0] * 0.5)
//
// Pure streaming op: 328 MB traffic, ~0.16 GFLOP -> bandwidth bound (~14 us at
// 23.3 TB/s). Strategy: b128 NT loads/stores (avoid thrashing the 192MB L2
// with a one-touch stream), wave32 ds_bpermute shuffle for the neighbor point
// (lane 0 of each wave does one extra 8B NT load), global_prefetch_b8 to
// warm L2 one chunk ahead. rot in [0, 0.5] rad -> hardware v_sin/v_cos via
// __sinf/__cosf, no ocml range-reduction blob.
//
// No WMMA by design: a 2x2 rotation maps to <1/8 useful lanes of
// V_WMMA_F32_16X16X4_F32 and scatters each point across 8 VGPRs of the D
// layout -- strictly a pessimization for a bandwidth-bound op.
// ---------------------------------------------------------------------------

typedef float v4f __attribute__((ext_vector_type(4)));
typedef float v2f __attribute__((ext_vector_type(2)));

#define NPTS    5000          // points per batch row
#define NPAIRS  2500          // float4 pairs per row (5000 is even)
#define TPB     256           // 8 waves of 32

__global__ __launch_bounds__(TPB)
void rotation_diff_kernel(const float* __restrict__ x,
                          const float* __restrict__ rand_value,
                          float* __restrict__ out)
{
    const int row  = blockIdx.y;                       // batch index b
    const int pair = blockIdx.x * TPB + threadIdx.x;   // pair index in row
    const bool active = pair < NPAIRS;
    const int  cpair  = active ? pair : (NPAIRS - 1);  // clamp for safe load

    // Uniform scalar angle; rot in [0, 0.5] rad -> hardware transcendentals
    // (v_cos_f32 / v_sin_f32, ~2 ULP in this range) are sufficient.
    const float rot = rand_value[0] * 0.5f;
    const float c = __cosf(rot);
    const float s = __sinf(rot);

    const size_t rowOff = (size_t)row * (NPTS * 2);
    const v4f* __restrict__ xin  = (const v4f*)(x   + rowOff);
    v4f*       __restrict__ xout = (v4f*)      (out + rowOff);

    // 16B NT load: two consecutive points (2n, 2n+1)
    v4f a = __builtin_nontemporal_load(&xin[cpair]);

    // Rotate both points (matches reference rotate-then-diff order)
    float r0x = c * a.x - s * a.y;
    float r0y = s * a.x + c * a.y;
    float r1x = c * a.z - s * a.w;
    float r1y = s * a.z + c * a.w;

    // Previous pair's rotated second point via wave32 shuffle (ds_bpermute).
    // Executed by all lanes so EXEC is full for the permute.
    float px = __shfl_up(r1x, 1, 32);
    float py = __shfl_up(r1y, 1, 32);

    const int lane = threadIdx.x & 31;
    if (lane == 0 && active) {
        if (pair == 0) {
            // n == 0: diff against itself -> exact zero (prepend semantics)
            px = r0x;
            py = r0y;
        } else {
            // Cross-wave boundary: one extra 8B load (same cacheline as the
            // previous wave's b128 load -> L2/L0 hit, no extra DRAM traffic)
            const v2f p = __builtin_nontemporal_load(
                (const v2f*)(x + rowOff + (size_t)(2 * pair - 1) * 2));
            px = c * p.x - s * p.y;
            py = s * p.x + c * p.y;
        }
    }

    if (active) {
        v4f o;
        o.x = r0x - px;   // point 2n
        o.y = r0y - py;
        o.z = r1x - r0x;  // point 2n+1
        o.w = r1y - r0y;
        __builtin_nontemporal_store(o, &xout[pair]);

        // Warm L2 one chunk ahead along this row's stream (global_prefetch_b8;
        // speculative, silently dropped if translation fails)
        const int pf = pair + TPB;
        if (pf < NPAIRS) {
            __builtin_prefetch((const void*)&xin[pf], 0, 1);
        }
    }
}

extern "C" void kernel_launch(void* const* d_in, const int* in_sizes, int n_in,
                              void* d_out, int out_size, void* d_ws, size_t ws_size,
                              hipStream_t stream)
{
    const float* x  = (const float*)d_in[0];   // [B, 5000, 2] f32
    const float* rv = (const float*)d_in[1];   // [1] f32
    float* out = (float*)d_out;                // [B, 5000, 2] f32

    const int rows = in_sizes[0] / (NPTS * 2); // B = 4096
    dim3 grid((NPAIRS + TPB - 1) / TPB, rows); // 10 x 4096 blocks
    rotation_diff_kernel<<<grid, TPB, 0, stream>>>(x, rv, out);
}